// SubNN_43782896615600
// MI455X (gfx1250) — compile-verified
//
#include <hip/hip_runtime.h>

// ---------------- problem constants (match reference) ----------------
#define NLAYERS 4
#define Hdim    256        // hidden width
#define Fdim    128        // Fourier features (output is 2*F = 256 wide)
#define DIMS    3
#define NROWS   4096       // coords rows; network processes 2*NROWS rows
#define TOTROWS (2 * NROWS)
#define ROWS    32         // rows per workgroup
#define STRIDE  260        // padded LDS row stride (260 % 64 == 4 -> no bank conflicts, 16B-aligned)
#define INV2PI  0.15915494309189535f

typedef float v2f __attribute__((ext_vector_type(2)));
typedef float v8f __attribute__((ext_vector_type(8)));

// v_sin_f32 / v_cos_f32 compute sin(2*pi*x) / cos(2*pi*x)
__device__ __forceinline__ float vsin_turns(float t) { return __builtin_amdgcn_sinf(t); }
__device__ __forceinline__ float vcos_turns(float t) { return __builtin_amdgcn_cosf(t); }
__device__ __forceinline__ float sin_rad(float x)    { return __builtin_amdgcn_sinf(x * INV2PI); }

// ---------------- phase 1: Fourier features ----------------
// dst[n][f]      = sin(2*pi * x_n . B[:,f])
// dst[n][128+f]  = cos(2*pi * x_n . B[:,f])
__device__ __forceinline__ void input_map(const float* __restrict__ coords,
                                          const float* __restrict__ Bm,
                                          float* __restrict__ dst,
                                          int row0, int tid)
{
    for (int i = tid; i < ROWS * Fdim; i += 256) {
        const int n = i >> 7;       // / 128
        const int f = i & (Fdim - 1);
        const int r = row0 + n;
        const float* cp = (r < NROWS) ? (coords + (size_t)r * 6)
                                      : (coords + (size_t)(r - NROWS) * 6 + DIMS);
        const float p = cp[0] * Bm[f] + cp[1] * Bm[Fdim + f] + cp[2] * Bm[2 * Fdim + f];
        dst[n * STRIDE + f]        = vsin_turns(p);   // sin(2*pi*p)
        dst[n * STRIDE + Fdim + f] = vcos_turns(p);   // cos(2*pi*p)
    }
}

// ---------------- phase 2: sin_layer ----------------
// dst[n][h] = sum_k sin(src[n][k] * w[h][k]);  w is [256][256] row-major.
// thread t owns column h = t; weight chunk held in regs pre-scaled by 1/2pi;
// activation reads are LDS broadcasts (same address across the wave).
__device__ __forceinline__ void sin_layer(const float* __restrict__ w,
                                          const float* __restrict__ src,
                                          float* __restrict__ dst, int tid)
{
    const int h = tid;
    float acc[ROWS];
#pragma unroll
    for (int n = 0; n < ROWS; ++n) acc[n] = 0.0f;

    for (int kc = 0; kc < Hdim; kc += 32) {
        float4 wv[8];
        const float4* wp = reinterpret_cast<const float4*>(w + (size_t)h * Hdim + kc);
#pragma unroll
        for (int q = 0; q < 8; ++q) {
            float4 t = wp[q];
            wv[q].x = t.x * INV2PI; wv[q].y = t.y * INV2PI;
            wv[q].z = t.z * INV2PI; wv[q].w = t.w * INV2PI;
        }
#pragma unroll 4
        for (int n = 0; n < ROWS; ++n) {
            const float4* a4 = reinterpret_cast<const float4*>(src + n * STRIDE + kc);
            float s = 0.0f;
#pragma unroll
            for (int q = 0; q < 8; ++q) {
                float4 a = a4[q];                       // ds_load_b128 broadcast
                s += vsin_turns(a.x * wv[q].x);
                s += vsin_turns(a.y * wv[q].y);
                s += vsin_turns(a.z * wv[q].z);
                s += vsin_turns(a.w * wv[q].w);
            }
            acc[n] += s;
        }
    }
#pragma unroll
    for (int n = 0; n < ROWS; ++n) dst[n * STRIDE + h] = acc[n];
}

// ---------------- phase 3: dense layer with WMMA fp32 ----------------
// dst[n][j] = sin(sum_k src[n][k] * w[j][k] + bias[j])
// 8 waves * 4 tiles = 32 (16x16) output tiles over (32 rows x 256 cols).
// K accumulated with V_WMMA_F32_16X16X4_F32.
__device__ __forceinline__ void dense_sin(const float* __restrict__ w,
                                          const float* __restrict__ bias,
                                          const float* __restrict__ src,
                                          float* __restrict__ dst, int tid)
{
    const int lane = tid & 31;
    const int wid  = tid >> 5;       // 0..7
    const int mt   = wid & 1;        // row-tile (0/1)
    const int half = lane >> 4;      // 0: lanes 0-15, 1: lanes 16-31
    const int Mr   = lane & 15;
    const int Nc   = lane & 15;
    const int ntBase = (wid >> 1) * 4;

    const float* arow = src + (mt * 16 + Mr) * STRIDE;

    for (int t = 0; t < 4; ++t) {
        const int col0 = (ntBase + t) * 16;
        const float* brow = w + (size_t)(col0 + Nc) * Hdim;
        v8f c = {0.f, 0.f, 0.f, 0.f, 0.f, 0.f, 0.f, 0.f};
#pragma unroll 8
        for (int k0 = 0; k0 < Hdim; k0 += 4) {
            // A 16x4: VGPR0 = K(0|2), VGPR1 = K(1|3) by half-wave
            const float2 av = *reinterpret_cast<const float2*>(arow + k0 + 2 * half);
            v2f a; a.x = av.x; a.y = av.y;
            // B 4x16: VGPR0 = K(0|1), VGPR1 = K(2|3) by half-wave
            const float4 bq = *reinterpret_cast<const float4*>(brow + k0);
            v2f b; b.x = half ? bq.y : bq.x; b.y = half ? bq.w : bq.z;
            c = __builtin_amdgcn_wmma_f32_16x16x4_f32(false, a, false, b,
                                                      (short)0, c, false, false);
        }
        const int   col  = col0 + Nc;
        const float bval = bias[col];
#pragma unroll
        for (int v = 0; v < 8; ++v) {
            const int Mrow = mt * 16 + v + half * 8;   // C/D layout
            dst[Mrow * STRIDE + col] = sin_rad(c[v] + bval);
        }
    }
}

// ---------------- fused network kernel ----------------
extern "C" __global__ __launch_bounds__(256)
void fused_siren(const float* __restrict__ coords, const float* __restrict__ Bm,
                 const float* __restrict__ w0,     const float* __restrict__ wsArr,
                 const float* __restrict__ w_last, const float* __restrict__ w1,
                 const float* __restrict__ b1,     float* __restrict__ out)
{
    __shared__ float bufA[ROWS * STRIDE];
    __shared__ float bufB[ROWS * STRIDE];
    const int tid  = threadIdx.x;
    const int row0 = blockIdx.x * ROWS;

    input_map(coords, Bm, bufA, row0, tid);
    __syncthreads();

    sin_layer(w0, bufA, bufB, tid);
    __syncthreads();
    dense_sin(w1, b1, bufB, bufA, tid);
    __syncthreads();

    for (int ii = 1; ii < NLAYERS; ++ii) {
        sin_layer(wsArr + (size_t)(ii - 1) * Hdim * Hdim, bufA, bufB, tid);
        __syncthreads();
        dense_sin(w1 + (size_t)ii * Hdim * Hdim, b1 + ii * Hdim, bufB, bufA, tid);
        __syncthreads();
    }

    sin_layer(w_last, bufA, bufB, tid);
    __syncthreads();

#pragma unroll
    for (int n = 0; n < ROWS; ++n)
        out[(size_t)(row0 + n) * Hdim + tid] = bufB[n * STRIDE + tid];
}

// ---------------- host entry ----------------
extern "C" void kernel_launch(void* const* d_in, const int* in_sizes, int n_in,
                              void* d_out, int out_size, void* d_ws, size_t ws_size,
                              hipStream_t stream)
{
    const float* coords = (const float*)d_in[0];   // (4096, 6)
    const float* Bm     = (const float*)d_in[1];   // (3, 128)
    const float* w0     = (const float*)d_in[2];   // (256, 256)
    const float* wsArr  = (const float*)d_in[3];   // (3, 256, 256)
    const float* w_last = (const float*)d_in[4];   // (256, 256)
    const float* w1     = (const float*)d_in[5];   // (4, 256, 256)
    const float* b1     = (const float*)d_in[6];   // (4, 256)
    float* out = (float*)d_out;                    // 8192*256 floats, then coords

    fused_siren<<<TOTROWS / ROWS, 256, 0, stream>>>(coords, Bm, w0, wsArr,
                                                    w_last, w1, b1, out);

    // second element of the returned tuple: coords passthrough
    hipMemcpyAsync(out + (size_t)TOTROWS * Hdim, d_in[0],
                   (size_t)NROWS * 2 * DIMS * sizeof(float),
                   hipMemcpyDeviceToDevice, stream);
}